// CTCLoss_85048942396174
// MI455X (gfx1250) — compile-verified
//
#include <hip/hip_runtime.h>
#include <cstdint>

#define DEVFN __device__ __forceinline__

constexpr int   kC           = 37;     // classes (36 labels + blank)
constexpr int   kBlank       = 36;
constexpr float kNeg         = -1e30f;
constexpr int   kTChunk      = 32;                       // timesteps staged per LDS chunk (== wave32)
constexpr int   kChunkFloats = kTChunk * kC;             // 1184 floats = 4736 B (16B-aligned stride)
constexpr int   kChunkB128   = (kChunkFloats * 4) / 16;  // 296 x 16B transfers
constexpr int   kWaves       = 4;                        // waves (samples) per block

typedef float v2f __attribute__((ext_vector_type(2)));
typedef float v8f __attribute__((ext_vector_type(8)));
typedef int   v4i_vs __attribute__((vector_size(16)));  // matches async builtin's pointee
typedef __attribute__((address_space(1))) v4i_vs gv4i;  // global 16B vector
typedef __attribute__((address_space(3))) v4i_vs lv4i;  // LDS 16B vector

#if defined(__has_builtin)
#if __has_builtin(__builtin_amdgcn_global_load_async_to_lds_b128)
#define HAVE_ASYNC_LDS 1
#endif
#endif

// log(exp(a)+exp(b)) -- safe with kNeg sentinels (exp underflows to 0)
DEVFN float lae2(float a, float b) {
  float m = fmaxf(a, b);
  return m + __logf(__expf(a - m) + __expf(b - m));
}
DEVFN float lae3(float a, float b, float c) {
  float m = fmaxf(fmaxf(a, b), c);
  return m + __logf(__expf(a - m) + __expf(b - m) + __expf(c - m));
}

// Stage one 32-timestep chunk (4736 B) of this sample's logits into LDS (ASYNCcnt path).
DEVFN void async_chunk_copy(const float* gsrc, float* ldst, int lane) {
#ifdef HAVE_ASYNC_LDS
  gv4i* g = (gv4i*)(uintptr_t)gsrc;
  lv4i* l = (lv4i*)(uint32_t)(uintptr_t)ldst;  // low 32 bits of generic LDS ptr = LDS offset
#pragma unroll
  for (int r = 0; r < (kChunkB128 + 31) / 32; ++r) {
    int idx = r * 32 + lane;
    if (idx < kChunkB128)
      __builtin_amdgcn_global_load_async_to_lds_b128(g + idx, l + idx, 0, 0);
  }
#else
  const float4* gs = (const float4*)gsrc;
  float4*       ld = (float4*)ldst;
  for (int i = lane; i < kChunkB128; i += 32) ld[i] = gs[i];
#endif
}

DEVFN void wait_copies() {
#ifdef HAVE_ASYNC_LDS
#if __has_builtin(__builtin_amdgcn_s_wait_asynccnt)
  __builtin_amdgcn_s_wait_asynccnt(0);
#else
  asm volatile("s_wait_asynccnt 0" ::: "memory");
#endif
#endif
  asm volatile("" ::: "memory");
}

// One wave per batch sample. Phase A (t-parallel): lane tt computes the full
// log-sum-exp of timestep t0+tt privately (no cross-lane reductions). Phase B
// (serial in t): lane s carries alpha[s]; alpha[32] is a replicated scalar.
// Logits stream global->LDS double-buffered with ASYNCcnt. All LDS accesses
// index the __shared__ arrays directly so they lower to ds_load/ds_store.
__global__ __launch_bounds__(kWaves * 32)
void ctc_alpha_kernel(const float* __restrict__ logits,
                      const int*   __restrict__ targets,
                      const int*   __restrict__ ilen,
                      const int*   __restrict__ tlen,
                      float*       __restrict__ loss,
                      int B, int T, int L)
{
  __shared__ __align__(16) float lbuf[kWaves][2][kChunkFloats]; // ~37.9 KB
  __shared__ float lseBuf[kWaves][kTChunk];
  __shared__ float lalpha[kWaves][40];

  const int lane = threadIdx.x & 31;
  const int wv   = threadIdx.x >> 5;
  const int b    = blockIdx.x * kWaves + wv;   // wave-uniform
  if (b >= B) return;

  const int iL = ilen[b];
  const int tL = tlen[b];

  // Extended label for state s=lane: even -> blank, odd -> tgt[(s-1)/2].
  int  extc = kBlank;
  bool skip = false;
  if (lane & 1) {
    extc = targets[(size_t)b * L + (lane >> 1)];
    if (lane >= 3)
      skip = (extc != targets[(size_t)b * L + (lane >> 1) - 1]);
  }
  const bool vS  = lane < 2 * tL + 1;
  const bool v32 = 32   < 2 * tL + 1;

  const float* gb = logits + (size_t)b * T * kC;

  async_chunk_copy(gb, &lbuf[wv][0][0], lane);

  float alpha = kNeg, a32 = kNeg, fin = kNeg;
  const int nchunk = T / kTChunk;              // T=256 -> 8 chunks
  for (int ck = 0; ck < nchunk; ++ck) {
    const int sel = ck & 1;
    wait_copies();                             // chunk ck resident in LDS
    if (ck + 1 < nchunk)                       // prefetch next chunk into the other buffer
      async_chunk_copy(gb + (size_t)(ck + 1) * kChunkFloats, &lbuf[wv][sel ^ 1][0], lane);

    // ---- Phase A: per-lane log-sum-exp for 32 timesteps (t-parallel) ----
    {
      float v[kC];
#pragma unroll
      for (int c = 0; c < kC; ++c)             // 37 conflict-free ds_load_b32
        v[c] = lbuf[wv][sel][lane * kC + c];
      float mmax = v[0];
#pragma unroll
      for (int c = 1; c < kC; ++c) mmax = fmaxf(mmax, v[c]);
      float ssum = 0.0f;
#pragma unroll
      for (int c = 0; c < kC; ++c) ssum += __expf(v[c] - mmax);
      lseBuf[wv][lane] = mmax + __logf(ssum);
    }

    // ---- Phase B: serial alpha recursion over the 32 staged timesteps ----
    for (int tt = 0; tt < kTChunk; ++tt) {
      const int t   = ck * kTChunk + tt;
      float lse = lseBuf[wv][tt];              // uniform-address broadcast ds load
      float lpS = lbuf[wv][sel][tt * kC + extc] - lse; // per-state lp gather (ds)
      float lpB = __shfl(lpS, 0, 32);          // lane 0 is a blank state -> lp(blank)

      if (t == 0) {
        alpha = (lane < 2 && vS) ? lpS : kNeg;
        a32   = kNeg;
      } else {
        float am1 = __shfl_up(alpha, 1, 32); if (lane < 1) am1 = kNeg;
        float am2 = __shfl_up(alpha, 2, 32); if (lane < 2) am2 = kNeg;
        float a31 = __shfl(alpha, 31, 32);
        float na  = lae3(alpha, am1, skip ? am2 : kNeg) + lpS;
        na = vS ? na : kNeg;
        float n32 = v32 ? (lae2(a32, a31) + lpB) : kNeg;
        alpha = na;
        a32   = n32;
      }

      // capture final log-likelihood at t+1 == input_length (wave-uniform)
      if (t + 1 == iL) {
        lalpha[wv][lane] = alpha;
        if (lane == 0) lalpha[wv][32] = a32;
        if (lane == 0) {
          float e1 = lalpha[wv][2 * tL];
          float e2 = lalpha[wv][2 * tL - 1];
          fin = lae2(e1, e2);
        }
      }
    }
  }
  if (lane == 0) loss[b] = -fin / (float)tL;
}

// Batch mean via chained V_WMMA_F32_16X16X4_F32: D = A*ones + C accumulates
// 64 losses per WMMA; every D column = per-row partial sums.
__global__ __launch_bounds__(32)
void reduce_mean_kernel(const float* __restrict__ loss, float* __restrict__ out, int B)
{
  const int lane = threadIdx.x;                // wave32, EXEC all ones for WMMA
  v8f acc = {};
  v2f bones = {1.0f, 1.0f};
  const int m  = lane & 15;                    // A-matrix row (lanes 0-15: K0/K1, 16-31: K2/K3)
  const int k0 = (lane >> 4) * 2;
  const int nfull = (B / 64) * 64;
  for (int it = 0; it < nfull; it += 64) {
    v2f a;
    a[0] = loss[it + m * 4 + k0 + 0];
    a[1] = loss[it + m * 4 + k0 + 1];
    acc = __builtin_amdgcn_wmma_f32_16x16x4_f32(false, a, false, bones,
                                                (short)0, acc, false, false);
  }
  float ssum = 0.0f;
#pragma unroll
  for (int i = 0; i < 8; ++i) ssum += acc[i];  // lane n: sums its 8 D rows
  float total = __shfl(ssum, 0, 32) + __shfl(ssum, 16, 32); // M=0..7 + M=8..15 at N=0
  if (lane == 0) {
    for (int i = nfull; i < B; ++i) total += loss[i];
    out[0] = total / (float)B;
  }
}

extern "C" void kernel_launch(void* const* d_in, const int* in_sizes, int n_in,
                              void* d_out, int out_size, void* d_ws, size_t ws_size,
                              hipStream_t stream)
{
  const float* logits  = (const float*)d_in[0];
  const int*   targets = (const int*)d_in[1];
  const int*   ilen    = (const int*)d_in[2];
  const int*   tlen    = (const int*)d_in[3];

  const int B = in_sizes[2];
  const int L = in_sizes[1] / B;
  const int T = in_sizes[0] / (B * kC);        // setup: T=256 (divisible by kTChunk)

  float* loss = (float*)d_ws;                  // B floats of scratch

  dim3 grid((B + kWaves - 1) / kWaves), block(kWaves * 32);
  ctc_alpha_kernel<<<grid, block, 0, stream>>>(logits, targets, ilen, tlen, loss, B, T, L);
  reduce_mean_kernel<<<1, 32, 0, stream>>>(loss, (float*)d_out, B);
}